// Transporter_20615843021558
// MI455X (gfx1250) — compile-verified
//
#include <hip/hip_runtime.h>

// ---------------------------------------------------------------------------
// Transporter: block-diagonal per-sample transport (P=400000 rows, width 320).
//   out[p, 0:64]     = x[p, 0:64]                              (copy region)
//   out[p, 64+2s+i]  = sum_j op1[p][i][j]*x[p, 64+2s+j]        (pair region 1)
//   out[p, 192+2s+i] = sum_j op2[p][i][j]*x[p, 192+2s+j]       (pair region 2)
//
// Roofline: ~1.04 GB streamed once, ~0.41 GFLOP -> bandwidth bound (~45us at
// 23.3 TB/s). WMMA inapplicable (operator varies per sample). CDNA5 paths
// used: wave32 b128 NT streaming for the transform regions, and the async
// global<->LDS engine (ASYNCcnt) for the copy region, which never needs VGPRs.
// ---------------------------------------------------------------------------

typedef float float4v __attribute__((ext_vector_type(4)));
typedef int   v4i     __attribute__((ext_vector_type(4)));

#if __has_builtin(__builtin_amdgcn_global_load_async_to_lds_b128) && \
    __has_builtin(__builtin_amdgcn_global_store_async_from_lds_b128)
#define HAS_ASYNC_LDS 1
#else
#define HAS_ASYNC_LDS 0
#endif

// Builtin params are int4 pointers: addrspace(1) for global, addrspace(3) for LDS.
static __device__ __forceinline__ __attribute__((address_space(1))) v4i*
as_global_v4i(const void* p) {
    return (__attribute__((address_space(1))) v4i*)(v4i*)p;
}
static __device__ __forceinline__ __attribute__((address_space(3))) v4i*
as_lds_v4i(void* p) {
    return (__attribute__((address_space(3))) v4i*)(v4i*)p;
}

static __device__ __forceinline__ void wait_asynccnt0() {
#if __has_builtin(__builtin_amdgcn_s_wait_asynccnt)
    __builtin_amdgcn_s_wait_asynccnt(0);
#else
    asm volatile("s_wait_asynccnt 0x0" ::: "memory");
#endif
}

constexpr int CPT = 4;  // copy chunks (float4) per thread

// ---- copy region: cols 0..63 of each row, via async DMA through LDS -------
__global__ __launch_bounds__(256)
void Transporter_copy_kernel(const float4v* __restrict__ x,
                             float4v* __restrict__ out,
                             unsigned P)
{
    const unsigned total = P * 16u;                       // 16 float4 per row
    const unsigned base  = blockIdx.x * (256u * CPT) + threadIdx.x;

#if HAS_ASYNC_LDS
    __shared__ float4v stage[256 * CPT];                  // 16 KB staging
#pragma unroll
    for (int k = 0; k < CPT; ++k) {
        const unsigned g = base + (unsigned)k * 256u;
        if (g < total) {
            const unsigned a = (g >> 4) * 80u + (g & 15u);
            __builtin_amdgcn_global_load_async_to_lds_b128(
                as_global_v4i(x + a),
                as_lds_v4i(&stage[threadIdx.x + k * 256]),
                /*offset=*/0, /*cpol=*/0);
        }
    }
    wait_asynccnt0();   // per-wave: async LDS writes complete before LDS reads
#pragma unroll
    for (int k = 0; k < CPT; ++k) {
        const unsigned g = base + (unsigned)k * 256u;
        if (g < total) {
            const unsigned a = (g >> 4) * 80u + (g & 15u);
            __builtin_amdgcn_global_store_async_from_lds_b128(
                as_global_v4i(out + a),
                as_lds_v4i(&stage[threadIdx.x + k * 256]),
                /*offset=*/0, /*cpol=*/0);
        }
    }
    // s_endpgm performs an implicit wait-idle on all counters incl. ASYNCcnt.
#else
    // Fallback: plain non-temporal b128 streaming copy.
#pragma unroll
    for (int k = 0; k < CPT; ++k) {
        const unsigned g = base + (unsigned)k * 256u;
        if (g < total) {
            const unsigned a = (g >> 4) * 80u + (g & 15u);
            float4v v = __builtin_nontemporal_load(&x[a]);
            __builtin_nontemporal_store(v, &out[a]);
        }
    }
#endif
}

// ---- pair regions: two 2x2 matvecs per float4, VALU FMA path --------------
__global__ __launch_bounds__(256)
void Transporter_pair_kernel(const float4v* __restrict__ x,
                             const float4v* __restrict__ ops, // float4 per (n,p): [m00,m01,m10,m11]
                             float4v* __restrict__ out,
                             unsigned P)
{
    const unsigned idx  = blockIdx.x * 256u + threadIdx.x;
    const unsigned half = P * 32u;                        // 32 float4 per row per group
    if (idx >= 2u * half) return;

    const bool     isOp1 = (idx < half);
    const unsigned j     = isOp1 ? idx : (idx - half);
    const unsigned row   = j >> 5;
    const unsigned c     = (isOp1 ? 16u : 48u) + (j & 31u);
    const unsigned a     = row * 80u + c;

    // operators flat: ((n*P+p)*2+i)*2+j -> one float4 per (n,p); broadcast in-wave
    const float4v m = ops[(isOp1 ? P : 2u * P) + row];

    float4v v = __builtin_nontemporal_load(&x[a]);
    float4v r;
    r.x = fmaf(m.y, v.y, m.x * v.x);
    r.y = fmaf(m.w, v.y, m.z * v.x);
    r.z = fmaf(m.y, v.w, m.x * v.z);
    r.w = fmaf(m.w, v.w, m.z * v.z);
    __builtin_nontemporal_store(r, &out[a]);
}

extern "C" void kernel_launch(void* const* d_in, const int* in_sizes, int n_in,
                              void* d_out, int out_size, void* d_ws, size_t ws_size,
                              hipStream_t stream) {
    (void)n_in; (void)out_size; (void)d_ws; (void)ws_size;

    const float4v* x   = (const float4v*)d_in[0];   // [P, 320] f32  -> [P*80] float4
    const float4v* ops = (const float4v*)d_in[1];   // [3, P, 2, 2] f32 -> [3*P] float4
    float4v*       out = (float4v*)d_out;           // [P, 320] f32

    const unsigned P = (unsigned)(in_sizes[0] / 320);   // 400000

    const unsigned copyChunks = P * 16u;
    const unsigned copyBlocks = (copyChunks + 256u * CPT - 1u) / (256u * CPT);
    Transporter_copy_kernel<<<copyBlocks, 256, 0, stream>>>(x, out, P);

    const unsigned pairChunks = P * 64u;
    const unsigned pairBlocks = (pairChunks + 255u) / 256u;
    Transporter_pair_kernel<<<pairBlocks, 256, 0, stream>>>(x, ops, out, P);
}